// EKFCell_9852654977787
// MI455X (gfx1250) — compile-verified
//
#include <hip/hip_runtime.h>

// CDNA5 (gfx1250) wave32 EKF cell: one batch per wave, all matrix products on
// V_WMMA_F32_16X16X4_F32 (exact fp32 precision), 4x4 SPD inverse on VALU.
// Branchless predicated loads + full-tile LDS staging (no divergent branches).
// B=262144, N=8, M=4.

typedef __attribute__((ext_vector_type(2))) float v2f;
typedef __attribute__((ext_vector_type(8))) float v8f;

#define LDP 17  // padded LDS leading dim (bank-conflict free)

__device__ __forceinline__ v8f wmma4(v2f a, v2f b, v8f c) {
  // D = A(16x4) * B(4x16) + C(16x16), fp32 everywhere.
  return __builtin_amdgcn_wmma_f32_16x16x4_f32(false, a, false, b, (short)0, c,
                                               false, false);
}

__device__ __forceinline__ void wsync() {
  // Wave-local ordering fence for LDS stage/reload (DS is in-order per wave;
  // this stops the compiler from reordering across it).
  __builtin_amdgcn_wave_barrier();
  asm volatile("" ::: "memory");
}

// Branchless predicated load: clamp index to 0 (always valid) and cndmask the
// result. Avoids exec-mask branches entirely.
__device__ __forceinline__ float ldz(const float* base, int idx, bool ok) {
  float v = base[ok ? idx : 0];
  return ok ? v : 0.0f;
}

// A-fragment from global: element (M,K), M = lane%16, K = kbase+v+(lane>=16?2:0).
__device__ __forceinline__ v2f loadA_g(const float* base, int sM, int sK,
                                       int Mmax, int Kmax, int kbase, int lane) {
  int M = lane & 15;
  int Kh = kbase + ((lane & 16) ? 2 : 0);
  v2f a;
  a.x = ldz(base, M * sM + (Kh + 0) * sK, M < Mmax && (Kh + 0) < Kmax);
  a.y = ldz(base, M * sM + (Kh + 1) * sK, M < Mmax && (Kh + 1) < Kmax);
  return a;
}

// B-fragment from global: element (K,N), N = lane%16, K = kbase+v+(lane>=16?2:0).
__device__ __forceinline__ v2f loadB_g(const float* base, int sK, int sN,
                                       int Kmax, int Nmax, int kbase, int lane) {
  int N = lane & 15;
  int Kh = kbase + ((lane & 16) ? 2 : 0);
  v2f b;
  b.x = ldz(base, (Kh + 0) * sK + N * sN, N < Nmax && (Kh + 0) < Kmax);
  b.y = ldz(base, (Kh + 1) * sK + N * sN, N < Nmax && (Kh + 1) < Kmax);
  return b;
}

// Lower-triangular (linearized row-major tril: idx = i(i+1)/2 + j, j<=i).
__device__ __forceinline__ v2f loadA_tril(const float* t, int n, int kbase, int lane) {
  int M = lane & 15;
  int Kh = kbase + ((lane & 16) ? 2 : 0);
  int tri = (M * (M + 1)) >> 1;
  v2f a;
  a.x = ldz(t, tri + Kh + 0, M < n && (Kh + 0) <= M);
  a.y = ldz(t, tri + Kh + 1, M < n && (Kh + 1) <= M);
  return a;
}

// B-fragment of L^T: (K,N) -> L[N][K].
__device__ __forceinline__ v2f loadB_trilT(const float* t, int n, int kbase, int lane) {
  int N = lane & 15;
  int Kh = kbase + ((lane & 16) ? 2 : 0);
  int tri = (N * (N + 1)) >> 1;
  v2f b;
  b.x = ldz(t, tri + Kh + 0, N < n && (Kh + 0) <= N);
  b.y = ldz(t, tri + Kh + 1, N < n && (Kh + 1) <= N);
  return b;
}

// Stage a full 16x16 C-layout tile into per-wave LDS (padded, branchless: all
// 32 lanes store; tile zeros are stored too, so reloads need no bounds checks).
__device__ __forceinline__ void storeTile(float* lds, v8f t, int lane) {
  wsync();
  int col = lane & 15;
  int rofs = (lane & 16) ? 8 : 0;
#pragma unroll
  for (int v = 0; v < 8; ++v) lds[(v + rofs) * LDP + col] = t[v];
  wsync();
}

// Unconditional fragment reloads from a staged tile.
__device__ __forceinline__ v2f ldsA(const float* lds, int kbase, int lane) {
  int M = lane & 15;
  int Kh = kbase + ((lane & 16) ? 2 : 0);
  v2f a;
  a.x = lds[M * LDP + Kh + 0];
  a.y = lds[M * LDP + Kh + 1];
  return a;
}
__device__ __forceinline__ v2f ldsB(const float* lds, int kbase, int lane) {
  int N = lane & 15;
  int Kh = kbase + ((lane & 16) ? 2 : 0);
  v2f b;
  b.x = lds[(Kh + 0) * LDP + N];
  b.y = lds[(Kh + 1) * LDP + N];
  return b;
}
__device__ __forceinline__ v2f ldsBT(const float* lds, int kbase, int lane) {
  int N = lane & 15;
  int Kh = kbase + ((lane & 16) ? 2 : 0);
  v2f b;
  b.x = lds[N * LDP + Kh + 0];
  b.y = lds[N * LDP + Kh + 1];
  return b;
}

__device__ __forceinline__ float pick4(float a, float b, float c, float d, int n) {
  float r = a;
  r = (n == 1) ? b : r;
  r = (n == 2) ? c : r;
  r = (n == 3) ? d : r;
  return r;
}

__global__ __launch_bounds__(256) void ekf_wmma_kernel(
    const float* __restrict__ meas, const float* __restrict__ state,
    const float* __restrict__ Pcov, const float* __restrict__ Fm,
    const float* __restrict__ Hm, const float* __restrict__ qn,
    const float* __restrict__ rn, float* __restrict__ out_pred,
    float* __restrict__ out_state, float* __restrict__ out_cov, int B) {
  __shared__ float stage[8][16 * LDP];  // one padded 16x16 fp32 stage per wave
  const int lane = threadIdx.x & 31;
  const int wv = threadIdx.x >> 5;
  const int b = blockIdx.x * 8 + wv;
  if (b >= B) return;  // uniform per wave; EXEC stays all-ones for WMMA
  float* my = &stage[wv][0];

  const float* Fb = Fm + (size_t)b * 64;
  const float* Pb = Pcov + (size_t)b * 64;
  const float* Hb = Hm + (size_t)b * 32;
  const float* qb = qn + (size_t)b * 36;
  const float* rb = rn + (size_t)b * 10;
  const float* sb = state + (size_t)b * 8;
  const float* zb = meas + (size_t)b * 4;

  const v8f vzero = {};
  const bool hi = (lane & 16) != 0;
  const bool c0 = (lane & 15) == 0;
  const int Koff = hi ? 2 : 0;

  // ---- acc = Q = Lq * Lq^T -------------------------------------------------
  v8f acc = wmma4(loadA_tril(qb, 8, 0, lane), loadB_trilT(qb, 8, 0, lane), vzero);
  acc = wmma4(loadA_tril(qb, 8, 4, lane), loadB_trilT(qb, 8, 4, lane), acc);

  // ---- T2 = P * F^T --------------------------------------------------------
  v8f t2 = wmma4(loadA_g(Pb, 8, 1, 8, 8, 0, lane),
                 loadB_g(Fb, 1, 8, 8, 8, 0, lane), vzero);
  t2 = wmma4(loadA_g(Pb, 8, 1, 8, 8, 4, lane),
             loadB_g(Fb, 1, 8, 8, 8, 4, lane), t2);

  v2f aF0 = loadA_g(Fb, 8, 1, 8, 8, 0, lane);  // F as A, reused 2x
  v2f aF1 = loadA_g(Fb, 8, 1, 8, 8, 4, lane);

  // ---- Pn = F * T2 + Q  (= F P F^T + Q) ------------------------------------
  storeTile(my, t2, lane);
  v8f Pn = wmma4(aF0, ldsB(my, 0, lane), acc);
  Pn = wmma4(aF1, ldsB(my, 4, lane), Pn);

  // ---- ns = F * s  (column-0 tile) -----------------------------------------
  float sv0 = sb[Koff + 0], sv1 = sb[Koff + 1];
  float sv2 = sb[Koff + 4], sv3 = sb[Koff + 5];
  v2f bs0; bs0.x = c0 ? sv0 : 0.0f; bs0.y = c0 ? sv1 : 0.0f;
  v2f bs1; bs1.x = c0 ? sv2 : 0.0f; bs1.y = c0 ? sv3 : 0.0f;
  v8f ns = wmma4(aF0, bs0, vzero);
  ns = wmma4(aF1, bs1, ns);

  // broadcast ns column to all lanes
  float n0 = __shfl(ns[0], 0, 32), n1 = __shfl(ns[1], 0, 32);
  float n2 = __shfl(ns[2], 0, 32), n3 = __shfl(ns[3], 0, 32);
  float n4 = __shfl(ns[4], 0, 32), n5 = __shfl(ns[5], 0, 32);
  float n6 = __shfl(ns[6], 0, 32), n7 = __shfl(ns[7], 0, 32);

  v2f aH0 = loadA_g(Hb, 8, 1, 4, 8, 0, lane);  // H as A, reused 2x
  v2f aH1 = loadA_g(Hb, 8, 1, 4, 8, 4, lane);

  // ---- pred = H * ns -------------------------------------------------------
  v2f bn0; bn0.x = c0 ? (hi ? n2 : n0) : 0.0f; bn0.y = c0 ? (hi ? n3 : n1) : 0.0f;
  v2f bn1; bn1.x = c0 ? (hi ? n6 : n4) : 0.0f; bn1.y = c0 ? (hi ? n7 : n5) : 0.0f;
  v8f pr = wmma4(aH0, bn0, vzero);
  pr = wmma4(aH1, bn1, pr);

  float p0 = __shfl(pr[0], 0, 32), p1 = __shfl(pr[1], 0, 32);
  float p2 = __shfl(pr[2], 0, 32), p3 = __shfl(pr[3], 0, 32);
  float r0 = zb[0] - p0, r1 = zb[1] - p1, r2 = zb[2] - p2, r3 = zb[3] - p3;

  // ---- PHt = Pn * H^T ------------------------------------------------------
  storeTile(my, Pn, lane);
  v2f aPn0 = ldsA(my, 0, lane);
  v2f aPn1 = ldsA(my, 4, lane);
  v8f pht = wmma4(aPn0, loadB_g(Hb, 1, 8, 8, 4, 0, lane), vzero);
  pht = wmma4(aPn1, loadB_g(Hb, 1, 8, 8, 4, 4, lane), pht);

  // ---- S = H * PHt + R  (= H Pn H^T + R) -----------------------------------
  v8f S = wmma4(loadA_tril(rb, 4, 0, lane), loadB_trilT(rb, 4, 0, lane), vzero);
  storeTile(my, pht, lane);
  v2f aPHt = ldsA(my, 0, lane);    // PHt as A (K=4; cols 4..7 staged as zeros)
  v2f bPHtT = ldsBT(my, 0, lane);  // PHt^T as B (K=4), kept for upd_cov
  S = wmma4(aH0, ldsB(my, 0, lane), S);
  S = wmma4(aH1, ldsB(my, 4, lane), S);

  // ---- broadcast S, invert 4x4 SPD per lane (Schur block form) -------------
  float s00 = __shfl(S[0], 0, 32), s01 = __shfl(S[0], 1, 32);
  float s02 = __shfl(S[0], 2, 32), s03 = __shfl(S[0], 3, 32);
  float s11 = __shfl(S[1], 1, 32), s12 = __shfl(S[1], 2, 32);
  float s13 = __shfl(S[1], 3, 32);
  float s22 = __shfl(S[2], 2, 32), s23 = __shfl(S[2], 3, 32);
  float s33 = __shfl(S[3], 3, 32);

  float detA = s00 * s11 - s01 * s01;
  float idA = 1.0f / detA;
  float ai00 = s11 * idA, ai01 = -s01 * idA, ai11 = s00 * idA;
  float e00 = ai00 * s02 + ai01 * s12, e01 = ai00 * s03 + ai01 * s13;
  float e10 = ai01 * s02 + ai11 * s12, e11 = ai01 * s03 + ai11 * s13;
  float sc00 = s22 - (s02 * e00 + s12 * e10);
  float sc01 = s23 - (s02 * e01 + s12 * e11);
  float sc11 = s33 - (s03 * e01 + s13 * e11);
  float idS = 1.0f / (sc00 * sc11 - sc01 * sc01);
  float g00 = sc11 * idS, g01 = -sc01 * idS, g11 = sc00 * idS;
  float x00 = e00 * g00 + e01 * g01, x01 = e00 * g01 + e01 * g11;
  float x10 = e10 * g00 + e11 * g01, x11 = e10 * g01 + e11 * g11;
  float i00 = ai00 + x00 * e00 + x01 * e01;
  float i01 = ai01 + x00 * e10 + x01 * e11;
  float i11 = ai11 + x10 * e10 + x11 * e11;
  float i02 = -x00, i03 = -x01, i12 = -x10, i13 = -x11;
  float i22 = g00, i23 = g01, i33 = g11;

  // Sinv as B fragment: (K,N) -> Sinv[K][N], N<4
  int N4 = lane & 15;
  float q0 = pick4(i00, i01, i02, i03, N4);
  float q1 = pick4(i01, i11, i12, i13, N4);
  float q2 = pick4(i02, i12, i22, i23, N4);
  float q3 = pick4(i03, i13, i23, i33, N4);
  bool nin = (N4 < 4);
  v2f bSi; bSi.x = nin ? (hi ? q2 : q0) : 0.0f; bSi.y = nin ? (hi ? q3 : q1) : 0.0f;

  // ---- kgain = PHt * Sinv --------------------------------------------------
  v8f kg = wmma4(aPHt, bSi, vzero);

  // ---- upd_state = ns + K * res --------------------------------------------
  storeTile(my, kg, lane);
  v2f aK = ldsA(my, 0, lane);  // kgain cols 4..7 staged as zeros
  v2f bres; bres.x = c0 ? (hi ? r2 : r0) : 0.0f; bres.y = c0 ? (hi ? r3 : r1) : 0.0f;
  v8f ust = wmma4(aK, bres, ns);

  // ---- upd_cov = Pn - K * PHt^T  (= (I - K H) Pn, Pn symmetric) ------------
  v2f aKn; aKn.x = -aK.x; aKn.y = -aK.y;
  v8f ucov = wmma4(aKn, bPHtT, Pn);

  // ---- outputs -------------------------------------------------------------
  if (lane < 4) out_pred[(size_t)b * 4 + lane] = pick4(p0, p1, p2, p3, lane);
  if (lane == 0) {
#pragma unroll
    for (int v = 0; v < 8; ++v) out_state[(size_t)b * 8 + v] = ust[v];
  }
  if (lane < 8) {
#pragma unroll
    for (int v = 0; v < 8; ++v) out_cov[(size_t)b * 64 + v * 8 + lane] = ucov[v];
  }
}

extern "C" void kernel_launch(void* const* d_in, const int* in_sizes, int n_in,
                              void* d_out, int out_size, void* d_ws, size_t ws_size,
                              hipStream_t stream) {
  const float* meas = (const float*)d_in[0];   // (B,4)
  const float* st   = (const float*)d_in[1];   // (B,8)
  const float* P    = (const float*)d_in[2];   // (B,8,8)
  const float* F    = (const float*)d_in[3];   // (B,8,8)
  const float* H    = (const float*)d_in[4];   // (B,4,8)
  const float* q    = (const float*)d_in[5];   // (B,36)
  const float* r    = (const float*)d_in[6];   // (B,10)
  int B = in_sizes[0] / 4;

  float* out = (float*)d_out;
  float* out_pred = out;                       // (B,4)
  float* out_state = out + (size_t)B * 4;      // (B,8)
  float* out_cov = out + (size_t)B * 12;       // (B,8,8)

  int blocks = (B + 7) / 8;  // 8 waves (= 8 batches) per 256-thread block
  ekf_wmma_kernel<<<blocks, 256, 0, stream>>>(meas, st, P, F, H, q, r, out_pred,
                                              out_state, out_cov, B);
}